// Recommender_13048110645352
// MI455X (gfx1250) — compile-verified
//
#include <hip/hip_runtime.h>
#include <math.h>

#define N_USERS  100000
#define N_ENT    200000
#define N_EDGES  1000000
#define NNZ_CNT  1000000
#define CDIM     64
#define NFAC     4
#define NRELM1   10
#define TEMP_F   0.2f

typedef float v2f __attribute__((ext_vector_type(2)));
typedef float v8f __attribute__((ext_vector_type(8)));

__device__ __forceinline__ void atomic_add_f32(float* p, float v) {
    __hip_atomic_fetch_add(p, v, __ATOMIC_RELAXED, __HIP_MEMORY_SCOPE_AGENT);
}

// ---------------- degree / inverse count ----------------
__global__ void deg_kernel(const int* __restrict__ head, float* __restrict__ cnt) {
    int e = blockIdx.x * blockDim.x + threadIdx.x;
    if (e < N_EDGES) atomic_add_f32(&cnt[head[e]], 1.0f);
}

__global__ void invcnt_kernel(float* __restrict__ cnt) {
    int i = blockIdx.x * blockDim.x + threadIdx.x;
    if (i < N_ENT) cnt[i] = 1.0f / fmaxf(cnt[i], 1.0f);
}

// ---------------- prep: disen_weight = softmax(att)@weight, cor scalar ----------------
__global__ void prep_kernel(const float* __restrict__ att,     // [NFAC, NRELM1]
                            const float* __restrict__ weight,  // [NRELM1, CDIM]
                            float* __restrict__ disen_w,       // [NFAC, CDIM]
                            float* __restrict__ cor_out) {
    int tid = threadIdx.x;  // 256 threads
    if (tid < NFAC * CDIM) {
        int f = tid / CDIM, c = tid % CDIM;
        float mx = -1e30f;
        for (int r = 0; r < NRELM1; ++r) mx = fmaxf(mx, att[f * NRELM1 + r]);
        float den = 0.f;
        float ex[NRELM1];
        for (int r = 0; r < NRELM1; ++r) { ex[r] = __expf(att[f * NRELM1 + r] - mx); den += ex[r]; }
        float acc = 0.f;
        for (int r = 0; r < NRELM1; ++r) acc += (ex[r] / den) * weight[r * CDIM + c];
        disen_w[f * CDIM + c] = acc;
    }
    if (tid == 0) {
        float cor = 0.f;
        for (int r = 0; r < NRELM1; ++r) {
            float nrm2 = 0.f;
            for (int f = 0; f < NFAC; ++f) { float v = att[f * NRELM1 + r]; nrm2 += v * v; }
            float n = sqrtf(nrm2);
            float pos = 0.f;
            for (int f = 0; f < NFAC; ++f) { float v = att[f * NRELM1 + r] / n; pos += v * v; }
            float ttl = 0.f;
            for (int r2 = 0; r2 < NRELM1; ++r2)
                for (int f = 0; f < NFAC; ++f)
                    ttl += att[f * NRELM1 + r] * att[f * NRELM1 + r2];
            cor += (ttl - pos) / TEMP_F;
        }
        *cor_out = cor;
    }
}

// ---------------- KG edge scatter: one wave32 per edge, 2 contiguous ch/lane ----------------
__global__ void kg_scatter(const int* __restrict__ head, const int* __restrict__ tail,
                           const int* __restrict__ etype, const float* __restrict__ weight,
                           const float* __restrict__ ent, float* __restrict__ agg) {
    int gtid = blockIdx.x * blockDim.x + threadIdx.x;
    int e = gtid >> 5;
    int lane = gtid & 31;
    if (e >= N_EDGES) return;
    int h = head[e], t = tail[e], r = etype[e] - 1;
    int c = lane * 2;
    v2f ev = *(const v2f*)(ent + (size_t)t * CDIM + c);
    v2f wv = *(const v2f*)(weight + r * CDIM + c);
    float* arow = agg + (size_t)h * CDIM + c;
    atomic_add_f32(&arow[0], ev.x * wv.x);
    atomic_add_f32(&arow[1], ev.y * wv.y);
}

// ---------------- interaction scatter: user_agg += val * entity[col] ----------------
__global__ void inter_scatter(const int* __restrict__ rows, const int* __restrict__ cols,
                              const float* __restrict__ vals, const float* __restrict__ ent,
                              float* __restrict__ uagg) {
    int gtid = blockIdx.x * blockDim.x + threadIdx.x;
    int i = gtid >> 5;
    int lane = gtid & 31;
    if (i >= NNZ_CNT) return;
    int u = rows[i], cidx = cols[i];
    float val = vals[i];
    int c = lane * 2;
    v2f ev = *(const v2f*)(ent + (size_t)cidx * CDIM + c);
    float* arow = uagg + (size_t)u * CDIM + c;
    atomic_add_f32(&arow[0], val * ev.x);
    atomic_add_f32(&arow[1], val * ev.y);
}

// ---------------- entity finalize: scale by inv_cnt, l2norm, residual ----------------
__global__ void ent_finalize(const float* __restrict__ agg, const float* __restrict__ invc,
                             float* __restrict__ cur, float* __restrict__ res) {
    int gtid = blockIdx.x * blockDim.x + threadIdx.x;
    int row = gtid >> 5;
    int lane = gtid & 31;
    if (row >= N_ENT) return;
    float ic = invc[row];
    int c = lane * 2;
    v2f v = *(const v2f*)(agg + (size_t)row * CDIM + c);
    v.x *= ic; v.y *= ic;
    float ss = v.x * v.x + v.y * v.y;
    for (int o = 16; o > 0; o >>= 1) ss += __shfl_xor(ss, o, 32);
    float sc = 1.0f / fmaxf(sqrtf(ss), 1e-12f);
    v.x *= sc; v.y *= sc;
    *(v2f*)(cur + (size_t)row * CDIM + c) = v;
    v2f rv = *(const v2f*)(res + (size_t)row * CDIM + c);
    rv.x += v.x; rv.y += v.y;
    *(v2f*)(res + (size_t)row * CDIM + c) = rv;
}

// ---------------- user finalize: WMMA scores -> softmax -> modulate -> l2norm ----------------
// block = 256 threads = 8 waves; each wave owns a 16-user tile (128 users / block).
__global__ void user_finalize(const float* __restrict__ ucur, const float* __restrict__ uagg,
                              const float* __restrict__ latent,  // [NFAC, CDIM]
                              const float* __restrict__ disen_w, // [NFAC, CDIM]
                              float* __restrict__ unew, float* __restrict__ ures) {
    __shared__ __align__(16) float s_lat[16 * CDIM];   // zero-padded B: rows 0..3 = latent
    __shared__ __align__(16) float s_dw[NFAC * CDIM];
    __shared__ float s_scores[128][NFAC];
    int tid = threadIdx.x;
    if (tid < NFAC * CDIM) s_dw[tid] = disen_w[tid];
#pragma unroll
    for (int i = tid; i < 16 * CDIM; i += 256)
        s_lat[i] = (i < NFAC * CDIM) ? latent[i] : 0.f;
    __syncthreads();

    int wave = tid >> 5, lane = tid & 31;
    int tile = blockIdx.x * 8 + wave;
    int m = lane & 15, hi = lane >> 4;

    if (tile * 16 < N_USERS) {  // wave-uniform guard: EXEC all-ones inside
        const float* arow = ucur + (size_t)(tile * 16 + m) * CDIM;
        const float* brow = s_lat + m * CDIM;  // B[k][n] = s_lat[n][k], n = lane&15 (padded)
        v8f acc = {0.f, 0.f, 0.f, 0.f, 0.f, 0.f, 0.f, 0.f};
#pragma unroll
        for (int kk = 0; kk < 16; ++kk) {
            int k0 = kk * 4 + hi * 2;  // per ISA 16x4 f32 A layout
            v2f a = *(const v2f*)(arow + k0);
            v2f b = *(const v2f*)(brow + k0);
            acc = __builtin_amdgcn_wmma_f32_16x16x4_f32(
                false, a, false, b, (short)0, acc, false, false);
        }
        // D layout: lane l, reg v -> D[v + 8*hi][l & 15]
        if (m < NFAC) {
#pragma unroll
            for (int v = 0; v < 8; ++v)
                s_scores[wave * 16 + v + 8 * hi][m] = acc[v];
        }
    }
    __syncthreads();

    // softmax over the 4 factors, one thread per user
    if (tid < 128) {
        int u = blockIdx.x * 128 + tid;
        if (u < N_USERS) {
            float l0 = s_scores[tid][0], l1 = s_scores[tid][1];
            float l2 = s_scores[tid][2], l3 = s_scores[tid][3];
            float mx = fmaxf(fmaxf(l0, l1), fmaxf(l2, l3));
            float e0 = __expf(l0 - mx), e1 = __expf(l1 - mx);
            float e2 = __expf(l2 - mx), e3 = __expf(l3 - mx);
            float inv = 1.0f / (e0 + e1 + e2 + e3);
            s_scores[tid][0] = e0 * inv; s_scores[tid][1] = e1 * inv;
            s_scores[tid][2] = e2 * inv; s_scores[tid][3] = e3 * inv;
        }
    }
    __syncthreads();

    // modulate + l2norm + residual: each wave sweeps rows wave, wave+8, ...
    int c = lane * 2;
    for (int rr = wave; rr < 128; rr += 8) {
        int u = blockIdx.x * 128 + rr;
        if (u >= N_USERS) break;  // wave-uniform
        float s0 = s_scores[rr][0], s1 = s_scores[rr][1];
        float s2 = s_scores[rr][2], s3 = s_scores[rr][3];
        float mod0 = 1.0f + s0 * s_dw[c] + s1 * s_dw[CDIM + c] +
                     s2 * s_dw[2 * CDIM + c] + s3 * s_dw[3 * CDIM + c];
        float mod1 = 1.0f + s0 * s_dw[c + 1] + s1 * s_dw[CDIM + c + 1] +
                     s2 * s_dw[2 * CDIM + c + 1] + s3 * s_dw[3 * CDIM + c + 1];
        v2f v = *(const v2f*)(uagg + (size_t)u * CDIM + c);
        v.x *= mod0; v.y *= mod1;
        float ss = v.x * v.x + v.y * v.y;
        for (int o = 16; o > 0; o >>= 1) ss += __shfl_xor(ss, o, 32);
        float sc = 1.0f / fmaxf(sqrtf(ss), 1e-12f);
        v.x *= sc; v.y *= sc;
        *(v2f*)(unew + (size_t)u * CDIM + c) = v;
        v2f rv = *(const v2f*)(ures + (size_t)u * CDIM + c);
        rv.x += v.x; rv.y += v.y;
        *(v2f*)(ures + (size_t)u * CDIM + c) = rv;
    }
}

extern "C" void kernel_launch(void* const* d_in, const int* in_sizes, int n_in,
                              void* d_out, int out_size, void* d_ws, size_t ws_size,
                              hipStream_t stream) {
    const float* user_emb   = (const float*)d_in[0];
    const float* entity_emb = (const float*)d_in[1];
    const float* latent_emb = (const float*)d_in[2];
    const float* weight     = (const float*)d_in[3];
    const float* disen_att  = (const float*)d_in[4];
    const int*   edge_index = (const int*)d_in[5];
    const int*   edge_type  = (const int*)d_in[6];
    const int*   inter_rows = (const int*)d_in[7];
    const int*   inter_cols = (const int*)d_in[8];
    const float* inter_vals = (const float*)d_in[9];

    float* out        = (float*)d_out;
    float* entity_res = out;
    float* user_res   = out + (size_t)N_ENT * CDIM;
    float* cor_out    = out + (size_t)N_ENT * CDIM + (size_t)N_USERS * CDIM;

    float* ws      = (float*)d_ws;
    float* ent_cur = ws;
    float* ent_agg = ent_cur + (size_t)N_ENT * CDIM;
    float* usr_cur = ent_agg + (size_t)N_ENT * CDIM;
    float* usr_agg = usr_cur + (size_t)N_USERS * CDIM;
    float* inv_cnt = usr_agg + (size_t)N_USERS * CDIM;
    float* disen_w = inv_cnt + N_ENT;

    const int* head = edge_index;            // edge_index[0, :]
    const int* tail = edge_index + N_EDGES;  // edge_index[1, :]

    hipMemcpyAsync(ent_cur,    entity_emb, sizeof(float) * (size_t)N_ENT * CDIM,   hipMemcpyDeviceToDevice, stream);
    hipMemcpyAsync(usr_cur,    user_emb,   sizeof(float) * (size_t)N_USERS * CDIM, hipMemcpyDeviceToDevice, stream);
    hipMemcpyAsync(entity_res, entity_emb, sizeof(float) * (size_t)N_ENT * CDIM,   hipMemcpyDeviceToDevice, stream);
    hipMemcpyAsync(user_res,   user_emb,   sizeof(float) * (size_t)N_USERS * CDIM, hipMemcpyDeviceToDevice, stream);
    hipMemsetAsync(inv_cnt, 0, sizeof(float) * N_ENT, stream);

    deg_kernel<<<(N_EDGES + 255) / 256, 256, 0, stream>>>(head, inv_cnt);
    invcnt_kernel<<<(N_ENT + 255) / 256, 256, 0, stream>>>(inv_cnt);
    prep_kernel<<<1, 256, 0, stream>>>(disen_att, weight, disen_w, cor_out);

    for (int hop = 0; hop < 2; ++hop) {
        hipMemsetAsync(ent_agg, 0, sizeof(float) * (size_t)N_ENT * CDIM, stream);
        hipMemsetAsync(usr_agg, 0, sizeof(float) * (size_t)N_USERS * CDIM, stream);

        kg_scatter<<<(N_EDGES * 32 + 255) / 256, 256, 0, stream>>>(
            head, tail, edge_type, weight, ent_cur, ent_agg);
        inter_scatter<<<(NNZ_CNT * 32 + 255) / 256, 256, 0, stream>>>(
            inter_rows, inter_cols, inter_vals, ent_cur, usr_agg);
        ent_finalize<<<(N_ENT * 32 + 255) / 256, 256, 0, stream>>>(
            ent_agg, inv_cnt, ent_cur, entity_res);
        user_finalize<<<(N_USERS + 127) / 128, 256, 0, stream>>>(
            usr_cur, usr_agg, latent_emb, disen_w, usr_cur, user_res);
    }
}